// TransformLayer_11570641895873
// MI455X (gfx1250) — compile-verified
//
#include <hip/hip_runtime.h>

typedef __attribute__((ext_vector_type(2))) float v2f;
typedef __attribute__((ext_vector_type(8))) float v8f;

#define BATCH  65536
#define NJ     24
#define NTRACK 6
#define TPB    256
#define WAVES  (TPB / 32)

// ---- helpers -------------------------------------------------------------

__device__ __forceinline__ void rpy2r(float r, float p, float y, float* R) {
  float cr = __cosf(r), sr = __sinf(r);
  float cp = __cosf(p), sp = __sinf(p);
  float cy = __cosf(y), sy = __sinf(y);
  R[0] = cy * cp; R[1] = cy * sp * sr - sy * cr; R[2] = cy * sp * cr + sy * sr;
  R[3] = sy * cp; R[4] = sy * sp * sr + cy * cr; R[5] = sy * sp * cr - cy * sr;
  R[6] = -sp;     R[7] = cp * sr;                R[8] = cp * cr;
}

// affine transform as t[0..8] = R row-major, t[9..11] = p
__device__ __forceinline__ void compose(float* __restrict__ c,
                                        const float* __restrict__ a,
                                        const float* __restrict__ b) {
#pragma unroll
  for (int i = 0; i < 3; ++i) {
    float a0 = a[3 * i], a1 = a[3 * i + 1], a2 = a[3 * i + 2];
#pragma unroll
    for (int k = 0; k < 3; ++k)
      c[3 * i + k] = a0 * b[k] + a1 * b[3 + k] + a2 * b[6 + k];
    c[9 + i] = a0 * b[9] + a1 * b[10] + a2 * b[11] + a[9 + i];
  }
}

__device__ __forceinline__ void store_se3(float* __restrict__ dst, const float* t) {
  float4* d4 = (float4*)dst;            // dst is 64B aligned -> full-cacheline write
  d4[0] = make_float4(t[0], t[1], t[2], t[9]);
  d4[1] = make_float4(t[3], t[4], t[5], t[10]);
  d4[2] = make_float4(t[6], t[7], t[8], t[11]);
  d4[3] = make_float4(0.f, 0.f, 0.f, 1.f);
}

// ---- kernel --------------------------------------------------------------

__global__ __launch_bounds__(TPB) void fk_kernel(
    const float* __restrict__ rev_q,  const float* __restrict__ pri_q,
    const float* __restrict__ p_off,  const float* __restrict__ rpy_off,
    const float* __restrict__ rev_ax, const float* __restrict__ pri_ax,
    const float* __restrict__ p_trk,  const float* __restrict__ rpy_trk,
    float* __restrict__ out_track, float* __restrict__ out_joint) {
  // Per-joint WMMA basis rows: row0 = I|a, row1 = K|Ka, row2 = K2|K2a (cols 12..15 = 0)
  __shared__ __align__(16) float sB[NJ][3][16];
  __shared__ __align__(16) float sTo[NJ][12];
  __shared__ __align__(16) float sTt[NJ][12];
  __shared__ v2f                 sCoef[WAVES][32];      // {sin q, 1-cos q} per lane
  __shared__ __align__(16) float sR[WAVES][32][16];     // D-matrix staging

  const int tid  = threadIdx.x;
  const int lane = tid & 31;
  const int wv   = tid >> 5;
  const int b    = blockIdx.x * TPB + tid;

  // ---- per-joint constants (threads 0..23) ----
  if (tid < NJ) {
    const int j = tid;
    float R[9];
    rpy2r(rpy_off[3 * j], rpy_off[3 * j + 1], rpy_off[3 * j + 2], R);
#pragma unroll
    for (int i = 0; i < 9; ++i) sTo[j][i] = R[i];
#pragma unroll
    for (int i = 0; i < 3; ++i) sTo[j][9 + i] = p_off[3 * j + i];

    rpy2r(rpy_trk[3 * j], rpy_trk[3 * j + 1], rpy_trk[3 * j + 2], R);
#pragma unroll
    for (int i = 0; i < 9; ++i) sTt[j][i] = R[i];
#pragma unroll
    for (int i = 0; i < 3; ++i) sTt[j][9 + i] = p_trk[3 * j + i];

    float ax = rev_ax[3 * j], ay = rev_ax[3 * j + 1], az = rev_ax[3 * j + 2];
    float inv = 1.0f / (sqrtf(ax * ax + ay * ay + az * az) + 1e-8f);
    ax *= inv; ay *= inv; az *= inv;
    float K[9]  = {0.f, -az, ay,  az, 0.f, -ax,  -ay, ax, 0.f};
    float K2[9];
#pragma unroll
    for (int i = 0; i < 3; ++i)
#pragma unroll
      for (int k = 0; k < 3; ++k)
        K2[3 * i + k] = K[3 * i] * K[k] + K[3 * i + 1] * K[3 + k] + K[3 * i + 2] * K[6 + k];

    float ap[3] = {pri_ax[3 * j], pri_ax[3 * j + 1], pri_ax[3 * j + 2]};
#pragma unroll
    for (int i = 0; i < 3; ++i) {
#pragma unroll
      for (int k = 0; k < 3; ++k) {
        int n = 3 * i + k;
        sB[j][0][n] = (i == k) ? 1.0f : 0.0f;
        sB[j][1][n] = K[n];
        sB[j][2][n] = K2[n];
      }
      sB[j][0][9 + i] = ap[i];
      sB[j][1][9 + i] = K[3 * i] * ap[0] + K[3 * i + 1] * ap[1] + K[3 * i + 2] * ap[2];
      sB[j][2][9 + i] = K2[3 * i] * ap[0] + K2[3 * i + 1] * ap[1] + K2[3 * i + 2] * ap[2];
    }
#pragma unroll
    for (int n = 12; n < 16; ++n) { sB[j][0][n] = 0.f; sB[j][1][n] = 0.f; sB[j][2][n] = 0.f; }
  }
  __syncthreads();

  float cur[12] = {1.f, 0.f, 0.f, 0.f, 1.f, 0.f, 0.f, 0.f, 1.f, 0.f, 0.f, 0.f};
  float* jout = out_joint + (size_t)b * NJ * 16;
  float* tout = out_track + (size_t)b * NTRACK * 16;
  const size_t qbase = (size_t)b * NJ;

  for (int j = 0; j < NJ; ++j) {
    // --- Rodrigues coefficients for this lane's batch element ---
    float q   = rev_q[qbase + j];
    float s   = __sinf(q);
    float omc = 1.0f - __cosf(q);
    sCoef[wv][lane] = (v2f){s, omc};
    __builtin_amdgcn_wave_barrier();

    // --- B operand (4x16 basis, shared per joint); row3 is zero ---
    v2f bv;
    if (lane < 16) { bv.x = sB[j][0][lane];      bv.y = sB[j][1][lane]; }
    else           { bv.x = sB[j][2][lane - 16]; bv.y = 0.0f; }

    // --- A operands: 16x4 coeff rows per the CDNA5 A layout ---
    v2f a0, a1;
    if (lane < 16) {
      a0.x = 1.0f;                    a0.y = sCoef[wv][lane].x;        // K0,K1 rows 0..15
      a1.x = 1.0f;                    a1.y = sCoef[wv][lane + 16].x;   // rows 16..31
    } else {
      a0.x = sCoef[wv][lane - 16].y;  a0.y = 0.0f;                     // K2,K3
      a1.x = sCoef[wv][lane].y;       a1.y = 0.0f;
    }

    v8f c0 = {0.f, 0.f, 0.f, 0.f, 0.f, 0.f, 0.f, 0.f};
    v8f c1 = {0.f, 0.f, 0.f, 0.f, 0.f, 0.f, 0.f, 0.f};
    // D(16x16) = coeffs(16x4) @ basis(4x16): cols 0..8 = R row-major, 9..11 = R@pri_axis
    c0 = __builtin_amdgcn_wmma_f32_16x16x4_f32(false, a0, false, bv, (short)0, c0, false, false);
    c1 = __builtin_amdgcn_wmma_f32_16x16x4_f32(false, a1, false, bv, (short)0, c1, false, false);

    // --- scatter D layout (VGPR v: row v+8*(lane>=16), col lane%16) into LDS ---
    const int col   = lane & 15;
    const int rbase = (lane < 16) ? 0 : 8;
    if (col < 12) {
#pragma unroll
      for (int v = 0; v < 8; ++v) {
        sR[wv][rbase + v][col]      = c0[v];
        sR[wv][16 + rbase + v][col] = c1[v];
      }
    }
    __builtin_amdgcn_wave_barrier();

    // --- gather this lane's R and v = R@pri_axis ---
    float4 g0 = *(const float4*)&sR[wv][lane][0];
    float4 g1 = *(const float4*)&sR[wv][lane][4];
    float4 g2 = *(const float4*)&sR[wv][lane][8];
    __builtin_amdgcn_wave_barrier();

    float pj = pri_q[qbase + j];
    float Tj[12];
    Tj[0] = g0.x; Tj[1] = g0.y; Tj[2] = g0.z;
    Tj[3] = g0.w; Tj[4] = g1.x; Tj[5] = g1.y;
    Tj[6] = g1.z; Tj[7] = g1.w; Tj[8] = g2.x;
    Tj[9] = pj * g2.y; Tj[10] = pj * g2.z; Tj[11] = pj * g2.w;

    // --- sequential chain (VALU) + full-cacheline stores ---
    float pre[12];
    compose(pre, cur, &sTo[j][0]);
    store_se3(jout + j * 16, pre);

    float post[12];
    compose(post, pre, Tj);

    if ((j & 3) == 3) {
      float trk[12];
      compose(trk, post, &sTt[j][0]);
      store_se3(tout + (j >> 2) * 16, trk);
    }
#pragma unroll
    for (int i = 0; i < 12; ++i) cur[i] = post[i];
  }
}

// ---- launcher ------------------------------------------------------------

extern "C" void kernel_launch(void* const* d_in, const int* in_sizes, int n_in,
                              void* d_out, int out_size, void* d_ws, size_t ws_size,
                              hipStream_t stream) {
  const float* rev_q   = (const float*)d_in[0];
  const float* pri_q   = (const float*)d_in[1];
  const float* p_off   = (const float*)d_in[2];
  const float* rpy_off = (const float*)d_in[3];
  const float* rev_ax  = (const float*)d_in[4];
  const float* pri_ax  = (const float*)d_in[5];
  const float* p_trk   = (const float*)d_in[6];
  const float* rpy_trk = (const float*)d_in[7];

  float* out_track = (float*)d_out;                               // (B,6,4,4) first
  float* out_joint = out_track + (size_t)BATCH * NTRACK * 16;     // then (B,24,4,4)

  hipLaunchKernelGGL(fk_kernel, dim3(BATCH / TPB), dim3(TPB), 0, stream,
                     rev_q, pri_q, p_off, rpy_off, rev_ax, pri_ax, p_trk, rpy_trk,
                     out_track, out_joint);
}